// GraphUpSample_12120397709982
// MI455X (gfx1250) — compile-verified
//
#include <hip/hip_runtime.h>

// GraphUpSample fused as: out_row[96] = x_row[48] * Wbig[48x96] + biasBig[96]
// using V_WMMA_F32_16X16X4_F32 (exact fp32). Memory-bound (~302MB -> ~13us @ 23.3TB/s).

typedef __attribute__((ext_vector_type(2))) float v2f;
typedef __attribute__((ext_vector_type(8))) float v8f;

// Net effect of the reference's aliased in-place column reorder: out[col] = z[SRC[col]]
__constant__ int SRC[32] = {0,2,4,6,8,10,12,14, 1,5,9,13, 3,11, 7, 15,
                            1,3,5,7,9,11,13,15, 3,7,11,15, 7,15,15, 31};

#define ROWS_PER_BLOCK 128   // 8 waves x 16 rows
#define XS_STRIDE      52    // 48 floats padded: 16B-aligned b128 stores, conflict-free A reads
#define WB_STRIDE      100   // 96 cols padded to decouple half-wave banks

__global__ __launch_bounds__(256)
void graph_upsample_wmma(const float* __restrict__ x,
                         const float* __restrict__ W,
                         const float* __restrict__ bias,
                         float* __restrict__ out)
{
    __shared__ __align__(16) float xs[ROWS_PER_BLOCK * XS_STRIDE]; // 26624 B
    __shared__ float wb[48 * WB_STRIDE];                           // 19200 B
    __shared__ float bs[96];

    const int tid = threadIdx.x;
    const long long rbase = (long long)blockIdx.x * ROWS_PER_BLOCK;

    // ---- Stage 128 x-rows (48 floats each, contiguous) into LDS, coalesced ----
    const float4* xsrc = (const float4*)(x + rbase * 48);
    #pragma unroll
    for (int i = 0; i < 6; ++i) {
        int ch  = tid + i * 256;        // 1536 float4 chunks total
        int f   = ch * 4;               // 48 % 4 == 0 -> chunk stays within one row
        int row = f / 48, col = f % 48;
        float4 v = xsrc[ch];
        *(float4*)&xs[row * XS_STRIDE + col] = v;  // 16B aligned (XS_STRIDE%4==0)
    }

    // ---- Build block-diagonal weight matrix Wbig[48][96] and biasBig[96] in LDS ----
    // k = cc*16 + nn ; j = c*32 + col ; out[c,col] uses node n'=SRC[col]>>1, m'=SRC[col]&1
    for (int idx = tid; idx < 48 * 96; idx += 256) {
        int k  = idx / 96, j = idx % 96;
        int cc = k >> 4,  nn = k & 15;
        int c  = j >> 5,  col = j & 31;
        int s  = SRC[col];
        int np = s >> 1,  mp = s & 1;
        float v = (nn == np) ? W[np * 18 + (c * 2 + mp) * 3 + cc] : 0.0f;
        wb[k * WB_STRIDE + j] = v;
    }
    if (tid < 96) {
        int c = tid >> 5, col = tid & 31;
        int s = SRC[col];
        bs[tid] = bias[(s >> 1) * 6 + c * 2 + (s & 1)];
    }
    __syncthreads();

    const int wave = tid >> 5;
    const int lane = tid & 31;
    const int half = lane >> 4;   // lanes 16..31 hold K+2 / M+8 halves
    const int mrow = lane & 15;

    // ---- A fragments: 16x4 f32 layout (VGPR0: K=0|K=2, VGPR1: K=1|K=3) ----
    v2f a[12];
    const int abase = (wave * 16 + mrow) * XS_STRIDE + 2 * half;
    #pragma unroll
    for (int t = 0; t < 12; ++t)
        a[t] = *(const v2f*)&xs[abase + 4 * t];     // ds_load_b64, 8B aligned

    v8f acc[6];
    #pragma unroll
    for (int j = 0; j < 6; ++j) acc[j] = (v8f){};

    // ---- 72 x v_wmma_f32_16x16x4_f32, interleaved over 6 N-tiles ----
    #pragma unroll
    for (int t = 0; t < 12; ++t) {
        const int kb = (4 * t + 2 * half) * WB_STRIDE;
        #pragma unroll
        for (int j = 0; j < 6; ++j) {
            v2f bf;
            bf.x = wb[kb + j * 16 + mrow];              // B row K, col N=j*16+mrow
            bf.y = wb[kb + WB_STRIDE + j * 16 + mrow];  // B row K+1
            acc[j] = __builtin_amdgcn_wmma_f32_16x16x4_f32(
                false, a[t], false, bf, (short)0, acc[j], false, false);
        }
    }

    // ---- Bias + store. C/D layout: VGPR r -> row (r + 8*half), col = mrow ----
    float* orow = out + (rbase + wave * 16 + 8 * half) * 96;
    #pragma unroll
    for (int j = 0; j < 6; ++j) {
        float bb = bs[j * 16 + mrow];
        #pragma unroll
        for (int r = 0; r < 8; ++r)
            orow[r * 96 + j * 16 + mrow] = acc[j][r] + bb;
    }
}

extern "C" void kernel_launch(void* const* d_in, const int* in_sizes, int n_in,
                              void* d_out, int out_size, void* d_ws, size_t ws_size,
                              hipStream_t stream) {
    const float* x = (const float*)d_in[0];   // [8192,64,3,16] f32
    const float* W = (const float*)d_in[1];   // [16,6,3] f32
    const float* b = (const float*)d_in[2];   // [16,6] f32
    float* out = (float*)d_out;               // [8192,64,3,32] f32

    int rows   = in_sizes[0] / 48;            // 524288
    int blocks = rows / ROWS_PER_BLOCK;       // 4096 (exact)
    graph_upsample_wmma<<<blocks, 256, 0, stream>>>(x, W, b, out);
}